// Indexer_87230785781900
// MI455X (gfx1250) — compile-verified
//
#include <hip/hip_runtime.h>

// ---- problem constants (from reference setup) ----
#define BB   2
#define SS   2048
#define HID  2048
#define NH   16
#define HD   128
#define TOPK 128

typedef __attribute__((ext_vector_type(2))) float v2f;
typedef __attribute__((ext_vector_type(8))) float v8f;

// Canonical k-swizzle for V_WMMA_F32_16X16X4_F32 operands:
// lane needs K = 4j + 2*half + {0,1}; grouping pairs by hsel=(k>>1)&1 makes
// each lane's required values contiguous:  pos = hsel*(W/2) + (k>>2)*2 + (k&1)

// =====================================================================
// Kernel 0: row-wise k-swizzle of a [R, 2048] f32 matrix (wk_w / wp_w)
// =====================================================================
__global__ __launch_bounds__(256) void wswz_kernel(
    const float* __restrict__ in, float* __restrict__ out)
{
  const long i4  = (long)blockIdx.x * 256 + threadIdx.x;
  const long e   = i4 << 2;
  const long row = e >> 11;                  // W = 2048
  const int  k   = (int)(e & 2047);
  const int  j   = k >> 2;
  const float4 v = *reinterpret_cast<const float4*>(in + e);
  float2 lo; lo.x = v.x; lo.y = v.y;         // hsel 0
  float2 hi; hi.x = v.z; hi.y = v.w;         // hsel 1
  *reinterpret_cast<float2*>(out + row * 2048 + 2 * j)        = lo;
  *reinterpret_cast<float2*>(out + row * 2048 + 1024 + 2 * j) = hi;
}

// =====================================================================
// Kernel 1: key projection  key_raw[r,d] = hidden[r,:]·wk_w[d,:] + wk_b[d]
// M=4096, N=128, K=2048, pure f32 WMMA (16x16x4), 512 WMMAs per tile.
// =====================================================================
__global__ __launch_bounds__(256) void projk_kernel(
    const float* __restrict__ hid, const float* __restrict__ wkwswz,
    const float* __restrict__ wkb, float* __restrict__ keyraw)
{
  __shared__ __align__(16) float As[16 * 36];  // 16 rows x 32 k (swizzled)
  const int mbase = blockIdx.x << 4;
  const int tid  = threadIdx.x;
  const int wave = tid >> 5, lane = tid & 31;
  const int n = lane & 15, half = lane >> 4;
  const int dcol = wave * 16 + n;
  const int sm = tid >> 4;                     // staging row
  const int sc = tid & 15;                     // staging pair index
  const int sLds = sm * 36 + (sc & 1) * 16 + (sc >> 1) * 2;
  const int sGk  = sc * 2;

  v8f acc = {};
  for (int kt = 0; kt < HID / 32; ++kt) {
    const int K0 = kt * 32;
    __syncthreads();
    *reinterpret_cast<float2*>(&As[sLds]) =
        *reinterpret_cast<const float2*>(&hid[(long)(mbase + sm) * HID + K0 + sGk]);
    __syncthreads();

    float4 a4[4], b4[4];
    const float4* ap = reinterpret_cast<const float4*>(&As[n * 36 + half * 16]);
    const float4* bp = reinterpret_cast<const float4*>(
        &wkwswz[(long)dcol * HID + half * 1024 + kt * 16]);
#pragma unroll
    for (int i = 0; i < 4; ++i) { a4[i] = ap[i]; b4[i] = bp[i]; }
#pragma unroll
    for (int jl = 0; jl < 8; ++jl) {
      v2f av, bv;
      if (jl & 1) { av[0] = a4[jl >> 1].z; av[1] = a4[jl >> 1].w;
                    bv[0] = b4[jl >> 1].z; bv[1] = b4[jl >> 1].w; }
      else        { av[0] = a4[jl >> 1].x; av[1] = a4[jl >> 1].y;
                    bv[0] = b4[jl >> 1].x; bv[1] = b4[jl >> 1].y; }
      acc = __builtin_amdgcn_wmma_f32_16x16x4_f32(
          false, av, false, bv, (short)0, acc, false, false);
    }
  }
  const float bias = wkb[dcol];
#pragma unroll
  for (int i = 0; i < 8; ++i) {
    int m = i + half * 8;
    keyraw[(long)(mbase + m) * HD + dcol] = acc[i] + bias;
  }
}

// =====================================================================
// Kernel 2: head-weight projection  w[r,h] = (hidden·wp_wᵀ + b)·NH^-0.5
// M=4096, N=16, K=2048.  One wave per M-tile; f32 WMMA.
// =====================================================================
__global__ __launch_bounds__(256) void projw_kernel(
    const float* __restrict__ hid, const float* __restrict__ wpwswz,
    const float* __restrict__ wpb, float* __restrict__ wts)
{
  const int tid  = threadIdx.x;
  const int wave = tid >> 5, lane = tid & 31;
  const int n = lane & 15, half = lane >> 4;
  const int mbase = (blockIdx.x * 8 + wave) << 4;

  v8f acc = {};
  for (int kt = 0; kt < HID / 32; ++kt) {
    const int K0 = kt * 32;
    float4 b4[4];
    const float4* bp = reinterpret_cast<const float4*>(
        &wpwswz[(long)n * HID + half * 1024 + kt * 16]);
#pragma unroll
    for (int i = 0; i < 4; ++i) b4[i] = bp[i];
#pragma unroll
    for (int jl = 0; jl < 8; ++jl) {
      const float2 af = *reinterpret_cast<const float2*>(
          &hid[(long)(mbase + n) * HID + K0 + jl * 4 + half * 2]);
      v2f av, bv;
      av[0] = af.x; av[1] = af.y;
      if (jl & 1) { bv[0] = b4[jl >> 1].z; bv[1] = b4[jl >> 1].w; }
      else        { bv[0] = b4[jl >> 1].x; bv[1] = b4[jl >> 1].y; }
      acc = __builtin_amdgcn_wmma_f32_16x16x4_f32(
          false, av, false, bv, (short)0, acc, false, false);
    }
  }
  const float bias = wpb[n];
#pragma unroll
  for (int i = 0; i < 8; ++i) {
    int m = i + half * 8;
    wts[(long)(mbase + m) * NH + n] = (acc[i] + bias) * 0.25f;  // 16^-0.5
  }
}

// =====================================================================
// Kernel 3: RoPE on keys + write K rows in swizzled layout (f32)
// =====================================================================
__global__ __launch_bounds__(256) void rope_kernel(
    const float* __restrict__ keyraw, const float* __restrict__ cosb,
    const float* __restrict__ sinb, float* __restrict__ kswz)
{
  const int i4  = blockIdx.x * 256 + threadIdx.x;   // BB*SS*HD/4 quads
  const int e   = i4 << 2;
  const int row = e >> 7;
  const int d   = e & 127;
  const float4 kv = *reinterpret_cast<const float4*>(keyraw + e);
  float4 pr;
  if (d < 64) {
    const float4 p = *reinterpret_cast<const float4*>(keyraw + row * HD + d + 64);
    pr.x = -p.x; pr.y = -p.y; pr.z = -p.z; pr.w = -p.w;
  } else {
    pr = *reinterpret_cast<const float4*>(keyraw + row * HD + d - 64);
  }
  const float4 c = *reinterpret_cast<const float4*>(cosb + e);
  const float4 s = *reinterpret_cast<const float4*>(sinb + e);
  const int j = d >> 2;
  float2 lo, hi;
  lo.x = kv.x * c.x + pr.x * s.x;  lo.y = kv.y * c.y + pr.y * s.y;   // hsel 0
  hi.x = kv.z * c.z + pr.z * s.z;  hi.y = kv.w * c.w + pr.w * s.w;   // hsel 1
  *reinterpret_cast<float2*>(kswz + row * HD + 2 * j)      = lo;
  *reinterpret_cast<float2*>(kswz + row * HD + 64 + 2 * j) = hi;
}

// =====================================================================
// Kernel 4 (hot): fused  sum_h relu(q_h·kᵀ)·w[s,h] + causal mask, f32 WMMA.
// Q tile (16 heads x 16 rows, f32) staged *and k-swizzled* via
// global_load_async_to_lds_b64 (one b64 chunk = one swizzle unit),
// drained with s_wait_asynccnt.  B-frags reused across all 16 heads.
// =====================================================================
__global__ __launch_bounds__(256) void scores_kernel(
    const float* __restrict__ q,       // [B,NH,S,HD] f32 (natural layout)
    const float* __restrict__ kswz,    // [B,S,HD] f32, k-swizzled rows
    const float* __restrict__ w,       // [B,S,NH]
    float* __restrict__ scores)        // [B,S,S]
{
  __shared__ __align__(16) float Qs[NH * 16 * 132];  // [h*16+m][swizzled d], pad 4
  __shared__ __align__(16) float Ws[NH * 16];        // [h][m]

  const int bsIdx  = blockIdx.x;
  const int b      = bsIdx >> 7;
  const int sbase  = (bsIdx & 127) << 4;
  const int tchunk = blockIdx.y;
  const int tid    = threadIdx.x;

  // ---- async stage + swizzle Q tile: 16384 b64 chunks, 64 per thread ----
  {
    const unsigned ldsQ = (unsigned)(unsigned long long)(void*)&Qs[0];
    const int c2 = tid & 63;                         // k-pair index in row
    const int r0 = tid >> 6;
    const unsigned ldsOffC = (unsigned)((((c2 & 1) * 64) + ((c2 >> 1) * 2)) * 4);
#pragma unroll
    for (int it = 0; it < 64; ++it) {
      const int row = r0 + it * 4;                   // 0..255 = h*16+m
      const int h = row >> 4, m = row & 15;
      const void* gp = (const void*)(q +
          ((((long)b * NH + h) * SS + sbase + m) * HD + 2 * c2));
      const unsigned lp = ldsQ + (unsigned)(row * 132 * 4) + ldsOffC;
      asm volatile("global_load_async_to_lds_b64 %0, %1, off"
                   :: "v"(lp), "v"(gp) : "memory");
    }
  }
  {
    const int h = tid >> 4, m = tid & 15;
    Ws[h * 16 + m] = w[((long)b * SS + (sbase + m)) * NH + h];
  }
  asm volatile("s_wait_asynccnt 0x0" ::: "memory");
  __syncthreads();

  const int wave = tid >> 5, lane = tid & 31;
  const int n = lane & 15, half = lane >> 4;

#pragma unroll 1
  for (int tt = 0; tt < 4; ++tt) {
    const int ttile = tchunk * 32 + wave * 4 + tt;
    const int tbase = ttile << 4;

    {   // prefetch next tile's K row for this lane
      const int tn = (ttile + 1 < 128) ? (ttile + 1) : 127;
      __builtin_prefetch(kswz + (((long)b * SS + (tn << 4) + n) * HD), 0, 1);
    }

    // B fragments: this lane's contiguous 64-of-128 swizzled K values
    float4 b4[16];
    {
      const float4* bp = reinterpret_cast<const float4*>(
          kswz + (((long)b * SS + tbase + n) * HD + half * 64));
#pragma unroll
      for (int i = 0; i < 16; ++i) b4[i] = bp[i];
    }

    v8f acc = {};
#pragma unroll 1
    for (int h = 0; h < NH; ++h) {
      float4 a4[16];
      const float4* ap = reinterpret_cast<const float4*>(
          &Qs[(h * 16 + n) * 132 + half * 64]);      // A row m == lane&15
#pragma unroll
      for (int i = 0; i < 16; ++i) a4[i] = ap[i];

      v8f t = {};
#pragma unroll
      for (int j = 0; j < 32; ++j) {
        v2f av, bv;
        if (j & 1) { av[0] = a4[j >> 1].z; av[1] = a4[j >> 1].w;
                     bv[0] = b4[j >> 1].z; bv[1] = b4[j >> 1].w; }
        else       { av[0] = a4[j >> 1].x; av[1] = a4[j >> 1].y;
                     bv[0] = b4[j >> 1].x; bv[1] = b4[j >> 1].y; }
        t = __builtin_amdgcn_wmma_f32_16x16x4_f32(
            false, av, false, bv, (short)0, t, false, false);
      }
      const float4* wp = reinterpret_cast<const float4*>(&Ws[h * 16 + half * 8]);
      const float4 w0 = wp[0], w1 = wp[1];
      acc[0] += fmaxf(t[0], 0.f) * w0.x;
      acc[1] += fmaxf(t[1], 0.f) * w0.y;
      acc[2] += fmaxf(t[2], 0.f) * w0.z;
      acc[3] += fmaxf(t[3], 0.f) * w0.w;
      acc[4] += fmaxf(t[4], 0.f) * w1.x;
      acc[5] += fmaxf(t[5], 0.f) * w1.y;
      acc[6] += fmaxf(t[6], 0.f) * w1.z;
      acc[7] += fmaxf(t[7], 0.f) * w1.w;
    }

#pragma unroll
    for (int i = 0; i < 8; ++i) {
      const int m  = i + half * 8;
      const int sg = sbase + m;
      const int tg = tbase + n;
      const float v = acc[i] + ((tg <= sg) ? 0.f : -1e9f);  // causal mask
      scores[((long)b * SS + sg) * SS + tg] = v;
    }
  }
}

// =====================================================================
// Kernel 5: top-128 per row, descending, lower index wins ties (jax top_k)
// =====================================================================
__global__ __launch_bounds__(256) void topk_kernel(
    const float* __restrict__ scores, int* __restrict__ topk)
{
  __shared__ float sv[SS];
  __shared__ float rv[256];
  __shared__ int   ri[256];
  const int row = blockIdx.x;
  const int tid = threadIdx.x;
  const float* srow = scores + (long)row * SS;
  for (int i = tid; i < SS; i += 256) sv[i] = srow[i];
  __syncthreads();

  for (int it = 0; it < TOPK; ++it) {
    float bv = -INFINITY; int bi = 0;
    for (int i = tid; i < SS; i += 256) {   // strict > keeps lowest index
      float v = sv[i];
      if (v > bv) { bv = v; bi = i; }
    }
    rv[tid] = bv; ri[tid] = bi;
    __syncthreads();
    for (int off = 128; off > 0; off >>= 1) {
      if (tid < off) {
        float ov = rv[tid + off]; int oi = ri[tid + off];
        if (ov > rv[tid] || (ov == rv[tid] && oi < ri[tid])) {
          rv[tid] = ov; ri[tid] = oi;
        }
      }
      __syncthreads();
    }
    if (tid == 0) {
      int wi = ri[0];
      topk[(long)row * TOPK + it] = wi;
      sv[wi] = -INFINITY;
    }
    __syncthreads();
  }
}

// =====================================================================
extern "C" void kernel_launch(void* const* d_in, const int* in_sizes, int n_in,
                              void* d_out, int out_size, void* d_ws, size_t ws_size,
                              hipStream_t stream) {
  const float* hid  = (const float*)d_in[0];  // hidden_states [B,S,H]
  const float* q    = (const float*)d_in[1];  // query_states [B,NH,S,HD]
  const float* cosb = (const float*)d_in[2];  // cos [B,S,HD]
  const float* sinb = (const float*)d_in[3];  // sin [B,S,HD]
  // d_in[4] = mask: computed analytically (tril -> 0 / -1e9), never read
  const float* wkw  = (const float*)d_in[5];  // [HD,H]
  const float* wkb  = (const float*)d_in[6];  // [HD]
  const float* wpw  = (const float*)d_in[7];  // [NH,H]
  const float* wpb  = (const float*)d_in[8];  // [NH]

  char* ws = (char*)d_ws;
  size_t off = 0;
  float* keyraw = (float*)(ws + off); off += (size_t)BB * SS * HD * 4;   // 2 MB
  float* kswz   = (float*)(ws + off); off += (size_t)BB * SS * HD * 4;   // 2 MB
  float* wts    = (float*)(ws + off); off += (size_t)BB * SS * NH * 4;   // 256 KB
  float* wkwswz = (float*)(ws + off); off += (size_t)HD * HID * 4;       // 1 MB
  float* wpwswz = (float*)(ws + off); off += (size_t)NH * HID * 4;       // 128 KB

  int*   topk   = (int*)d_out;                                  // [B,1,S,128]
  float* scores = (float*)d_out + (size_t)BB * SS * TOPK;       // [B,1,S,S]

  wswz_kernel<<<(HD * HID) / (4 * 256), 256, 0, stream>>>(wkw, wkwswz);
  wswz_kernel<<<(NH * HID) / (4 * 256), 256, 0, stream>>>(wpw, wpwswz);
  projk_kernel<<<(BB * SS) / 16, 256, 0, stream>>>(hid, wkwswz, wkb, keyraw);
  projw_kernel<<<(BB * SS) / (16 * 8), 256, 0, stream>>>(hid, wpwswz, wpb, wts);
  rope_kernel<<<(BB * SS * HD) / (4 * 256), 256, 0, stream>>>(keyraw, cosb, sinb, kswz);
  scores_kernel<<<dim3((BB * SS) / 16, 4), 256, 0, stream>>>(q, kswz, wts, scores);
  topk_kernel<<<BB * SS, 256, 0, stream>>>(scores, topk);
}